// OuterProduct_58609123721696
// MI455X (gfx1250) — compile-verified
//
#include <hip/hip_runtime.h>

typedef float v2f __attribute__((ext_vector_type(2)));
typedef float v8f __attribute__((ext_vector_type(8)));

#define B_ 2
#define N_ 512
#define D_ 512
#define H_ 8
#define E_ 128

// Kernel 1: o12[b*N+n][h] = dot(x[b,n,:], W12[h,:]) + b12[h]   (B*N*16 outputs)
__global__ void k_o12(const float* __restrict__ x, const float* __restrict__ W12,
                      const float* __restrict__ b12, float* __restrict__ o12) {
    int idx = blockIdx.x * blockDim.x + threadIdx.x;   // 0..16383
    int row = idx >> 4;                                // b*N + n
    int h   = idx & 15;
    const float4* xr = (const float4*)(x + (size_t)row * D_);
    const float4* wr = (const float4*)(W12 + (size_t)h * D_);
    float acc = 0.f;
#pragma unroll 4
    for (int d4 = 0; d4 < D_ / 4; ++d4) {
        float4 xv = xr[d4], wv = wr[d4];
        acc += xv.x * wv.x + xv.y * wv.y + xv.z * wv.z + xv.w * wv.w;
    }
    o12[idx] = acc + b12[h];
}

// Kernel 2: v[((b*N+m)*E + e)*8 + i] = sum_j o2[b,m,j] * W3[e, i*8+j]
__global__ void k_v(const float* __restrict__ o12, const float* __restrict__ W3,
                    float* __restrict__ v) {
    int idx = blockIdx.x * blockDim.x + threadIdx.x;   // 0..1048575
    int i  = idx & 7;
    int e  = (idx >> 3) & (E_ - 1);
    int bm = idx >> 10;                                // b*N + m
    const float* o2 = o12 + bm * 16 + 8;
    const float* w  = W3 + e * (H_ * H_) + i * H_;
    float acc = 0.f;
#pragma unroll
    for (int j = 0; j < 8; ++j) acc += o2[j] * w[j];
    v[idx] = acc;
}

// Kernel 3: out[b,n,m,e] = sum_i o1[b,n,i] * v[b,m,e,i] + b3[e]
// One wave = one 16(n) x 16(e) tile via two V_WMMA_F32_16X16X4_F32 (K=8).
// Block = 256 threads = 8 waves = the 8 e-tiles of one (b, n-tile, m).
__global__ void __launch_bounds__(256) k_out(const float* __restrict__ o12,
                                             const float* __restrict__ v,
                                             const float* __restrict__ b3,
                                             float* __restrict__ out) {
    int lane = threadIdx.x & 31;
    int wave = threadIdx.x >> 5;            // e-tile 0..7
    int blk  = blockIdx.x;                  // 0..32767
    int m    = blk & (N_ - 1);
    int nt   = (blk >> 9) & 31;
    int b    = blk >> 14;
    int n0   = nt * 16;
    int e0   = wave * 16;

    int row  = lane & 15;                   // M index / N(col) index
    int half = lane >> 4;                   // 0: K=0,1   1: K=2,3
    int k0   = half * 2;

    // A fragment: o1[b, n0+row, k] ; o1 = o12[..., 0:8]
    const float* o1r = o12 + ((size_t)(b * N_) + n0 + row) * 16;
    v2f a0 = *(const v2f*)(o1r + k0);       // k = k0, k0+1
    v2f a1 = *(const v2f*)(o1r + 4 + k0);   // k = 4+k0, 5+k0

    // B fragment: B[k][col] = v[b, m, e0+col, k]
    const float* vr = v + (((size_t)(b * N_) + m) * E_ + e0 + row) * H_;
    v2f bv0 = *(const v2f*)(vr + k0);
    v2f bv1 = *(const v2f*)(vr + 4 + k0);

    v8f c = {};
    c = __builtin_amdgcn_wmma_f32_16x16x4_f32(false, a0, false, bv0,
                                              (short)0, c, false, false);
    c = __builtin_amdgcn_wmma_f32_16x16x4_f32(false, a1, false, bv1,
                                              (short)0, c, false, false);

    // C/D layout: VGPR r -> row M = r + half*8, col = lane&15 (e dim, contiguous)
    float bias = b3[e0 + row];
    size_t rstride = (size_t)N_ * E_;
    size_t base = (((size_t)(b * N_) + n0 + half * 8) * N_ + m) * E_ + e0 + row;
#pragma unroll
    for (int r = 0; r < 8; ++r) {
        // 256 MB streaming output, never re-read: non-temporal store
        __builtin_nontemporal_store(c[r] + bias, out + base + (size_t)r * rstride);
    }
}

extern "C" void kernel_launch(void* const* d_in, const int* in_sizes, int n_in,
                              void* d_out, int out_size, void* d_ws, size_t ws_size,
                              hipStream_t stream) {
    const float* x   = (const float*)d_in[0];
    const float* W12 = (const float*)d_in[1];
    const float* b12 = (const float*)d_in[2];
    const float* W3  = (const float*)d_in[3];
    const float* b3  = (const float*)d_in[4];
    float* out = (float*)d_out;

    float* o12 = (float*)d_ws;                 // B*N*16   = 16384 floats (64 KB)
    float* v   = o12 + (size_t)B_ * N_ * 16;   // B*N*E*H  = 1048576 floats (4 MB)

    k_o12<<<(B_ * N_ * 16) / 256, 256, 0, stream>>>(x, W12, b12, o12);
    k_v<<<(B_ * N_ * E_ * H_) / 256, 256, 0, stream>>>(o12, W3, v);
    k_out<<<B_ * (N_ / 16) * N_ * (E_ / 16) / 8, 256, 0, stream>>>(o12, v, b3, out);
}